// Attention_47519518163089
// MI455X (gfx1250) — compile-verified
//
#include <hip/hip_runtime.h>
#include <hip/hip_bf16.h>

typedef __attribute__((ext_vector_type(2))) float v2f;
typedef __attribute__((ext_vector_type(8))) float v8f;

#define WMMA_F32(A_, B_, C_) \
  __builtin_amdgcn_wmma_f32_16x16x4_f32(false, (A_), false, (B_), (short)0, (C_), false, false)

// Async memory->LDS copy (CDNA5, tracked by ASYNCcnt). lds_off is the wave-
// relative LDS byte offset (== low 32 bits of the generic address of a
// __shared__ object, per the LDS aperture mapping).
__device__ __forceinline__ void async_copy_b128(unsigned int lds_off, const float* gaddr)
{
    asm volatile("global_load_async_to_lds_b128 %0, %1, off"
                 :: "v"(lds_off), "v"(gaddr) : "memory");
}
__device__ __forceinline__ void wait_async0()
{
    asm volatile("s_wait_asynccnt 0x0" ::: "memory");
}
__device__ __forceinline__ unsigned int lds_offset(const void* p)
{
    return (unsigned int)(unsigned long long)p;   // low 32 bits = LDS offset
}

// Problem constants
static constexpr int kB  = 16;
static constexpr int kN  = 784;
static constexpr int kC  = 384;
static constexpr int kH  = 8;
static constexpr int kHD = 48;      // kC / kH
static constexpr int kIC = 1152;    // 3*kC
static constexpr int kRows = kB * kN;          // 12544
static constexpr int kQT = 112;                // query rows per attention block (7 waves * 16)
static constexpr int kKT = kN / 16;            // 49 key tiles

// ---------------------------------------------------------------------------
// GEMM: C[M,N] = A[M,K] @ B[N,K]^T   (fp32, WMMA f32 16x16x4)
// block = 256 threads (8 waves), tile = 128(M) x 64(N), K multiple of 16.
// Double-buffered LDS staging via GLOBAL_LOAD_ASYNC_TO_LDS_B128.
// ---------------------------------------------------------------------------
__global__ __launch_bounds__(256)
void gemm_nt_f32(const float* __restrict__ A, const float* __restrict__ Bm,
                 float* __restrict__ C, int M, int Nc, int K)
{
    __shared__ __align__(16) float lA[2][128 * 16];
    __shared__ __align__(16) float lB[2][64 * 16];

    const int tid   = threadIdx.x;
    const int wave  = tid >> 5;
    const int lane  = tid & 31;
    const int lhalf = lane & 15;
    const int lhi   = lane >> 4;
    const int m0 = blockIdx.x * 128;
    const int n0 = blockIdx.y * 64;

    // per-thread staging assignment: row = tid>>2 (0..63), 16B column chunk
    const int sr  = tid >> 2;
    const int sc4 = (tid & 3) << 2;

    v8f acc0 = {}, acc1 = {}, acc2 = {}, acc3 = {};

    auto issue_tile = [&](int buf, int k0) {
        async_copy_b128(lds_offset(&lA[buf][sr * 16 + sc4]),
                        &A[(size_t)(m0 + sr) * K + k0 + sc4]);
        async_copy_b128(lds_offset(&lA[buf][(sr + 64) * 16 + sc4]),
                        &A[(size_t)(m0 + sr + 64) * K + k0 + sc4]);
        async_copy_b128(lds_offset(&lB[buf][sr * 16 + sc4]),
                        &Bm[(size_t)(n0 + sr) * K + k0 + sc4]);
    };

    issue_tile(0, 0);

    int buf = 0;
    for (int k0 = 0; k0 < K; k0 += 16, buf ^= 1) {
        wait_async0();          // own async fills of lA/lB[buf] complete
        __syncthreads();        // everyone's fills complete; prev reads done
        if (k0 + 16 < K) issue_tile(buf ^ 1, k0 + 16);

        const float* __restrict__ cA = &lA[buf][0];
        const float* __restrict__ cB = &lB[buf][0];
        const int am = wave * 16 + lhalf;
        #pragma unroll
        for (int kk = 0; kk < 16; kk += 4) {
            // A fragment (16x4): lane<16 -> K=kk,kk+1 ; lane>=16 -> K=kk+2,kk+3
            v2f a  = *(const v2f*)&cA[am * 16 + kk + 2 * lhi];
            v2f b0 = *(const v2f*)&cB[(0 * 16 + lhalf) * 16 + kk + 2 * lhi];
            v2f b1 = *(const v2f*)&cB[(1 * 16 + lhalf) * 16 + kk + 2 * lhi];
            v2f b2 = *(const v2f*)&cB[(2 * 16 + lhalf) * 16 + kk + 2 * lhi];
            v2f b3 = *(const v2f*)&cB[(3 * 16 + lhalf) * 16 + kk + 2 * lhi];
            acc0 = WMMA_F32(a, b0, acc0);
            acc1 = WMMA_F32(a, b1, acc1);
            acc2 = WMMA_F32(a, b2, acc2);
            acc3 = WMMA_F32(a, b3, acc3);
        }
        __syncthreads();        // reads of buf done before it is refilled
    }

    // C layout: vgpr r -> M = r (lanes 0-15) / r+8 (lanes 16-31), N = lane%16
    #pragma unroll
    for (int r = 0; r < 8; ++r) {
        int mm = m0 + wave * 16 + r + 8 * lhi;
        size_t rowb = (size_t)mm * Nc + n0 + lhalf;
        C[rowb + 0]  = acc0[r];
        C[rowb + 16] = acc1[r];
        C[rowb + 32] = acc2[r];
        C[rowb + 48] = acc3[r];
    }
}

// ---------------------------------------------------------------------------
// BatchNorm statistics: partial sums per row-chunk, then finalize.
// ---------------------------------------------------------------------------
__global__ __launch_bounds__(128)
void colstats_partial(const float* __restrict__ Y, float* __restrict__ part,
                      int rows, int nc, int rowsPerChunk)
{
    int c = blockIdx.x * 128 + threadIdx.x;
    int chunk = blockIdx.y;
    int r0 = chunk * rowsPerChunk;
    int r1 = r0 + rowsPerChunk;
    if (r1 > rows) r1 = rows;
    float s = 0.f, ss = 0.f;
    for (int r = r0; r < r1; ++r) {
        float v = Y[(size_t)r * nc + c];
        s += v;
        ss += v * v;
    }
    part[((size_t)(2 * chunk)) * nc + c]     = s;
    part[((size_t)(2 * chunk + 1)) * nc + c] = ss;
}

__global__ __launch_bounds__(256)
void colstats_final(const float* __restrict__ part,
                    const float* __restrict__ gamma, const float* __restrict__ beta,
                    float* __restrict__ scale, float* __restrict__ shift,
                    int nc, int nchunk, float invRows)
{
    int c = blockIdx.x * 256 + threadIdx.x;
    if (c >= nc) return;
    float s = 0.f, ss = 0.f;
    for (int i = 0; i < nchunk; ++i) {
        s  += part[((size_t)(2 * i)) * nc + c];
        ss += part[((size_t)(2 * i + 1)) * nc + c];
    }
    float m   = s * invRows;
    float var = ss * invRows - m * m;
    float rstd = rsqrtf(var + 1e-5f);
    float g = gamma[c] * rstd;
    scale[c] = g;
    shift[c] = beta[c] - g * m;
}

// ---------------------------------------------------------------------------
// Bias expansion: Bg[h][n][m] = biases[h][idx[n][m]]  (gather once, reuse B=16x)
// ---------------------------------------------------------------------------
__global__ __launch_bounds__(256)
void bias_expand(const float* __restrict__ biases, const int* __restrict__ idx,
                 float* __restrict__ Bg)
{
    int n = blockIdx.x;
    int h = blockIdx.y;
    const float* brow = biases + (size_t)h * kN;
    const int*   irow = idx    + (size_t)n * kN;
    float*       orow = Bg + ((size_t)h * kN + n) * kN;
    for (int m = threadIdx.x; m < kN; m += 256)
        orow[m] = brow[irow[m]];
}

// ---------------------------------------------------------------------------
// Fused flash attention per (b, h, 112-query tile). 7 waves, one 16-row
// S/O stripe per wave. BN of q/k/v fused at load; hardswish fused at store.
// ---------------------------------------------------------------------------
__global__ __launch_bounds__(224)
void attention_fused(const float* __restrict__ Y,
                     const float* __restrict__ scale1, const float* __restrict__ shift1,
                     const float* __restrict__ Bg,
                     float* __restrict__ Ohs, float hdScale)
{
    __shared__ __align__(16) float lQ[kQT * kHD];   // 21504 B
    __shared__ __align__(16) float lK[16 * kHD];    //  3072 B
    __shared__ __align__(16) float lV[16 * kHD];    //  3072 B
    __shared__ __align__(16) float lP[7][16 * 16];  //  7168 B

    const int tid   = threadIdx.x;
    const int wave  = tid >> 5;
    const int lane  = tid & 31;
    const int lhalf = lane & 15;
    const int lhi   = lane >> 4;
    const int qbase = blockIdx.x * kQT;
    const int h     = blockIdx.y;
    const int b     = blockIdx.z;

    // Stage Q tile with BN + hd^-0.5 fused
    for (int i = tid; i < kQT * kHD; i += 224) {
        int r = i / kHD, d = i - r * kHD;
        int col = h * kHD + d;
        float y = Y[((size_t)b * kN + qbase + r) * kIC + col];
        lQ[i] = (y * scale1[col] + shift1[col]) * hdScale;
    }

    float m_run[8], l_run[8];
    #pragma unroll
    for (int r = 0; r < 8; ++r) { m_run[r] = -3.0e38f; l_run[r] = 0.f; }
    v8f o0 = {}, o1 = {}, o2 = {};

    const float* bg = Bg + ((size_t)h * kN) * kN;

    for (int kt = 0; kt < kKT; ++kt) {
        __syncthreads();
        // Stage one K/V tile (16 x 48) with BN fused
        for (int i = tid; i < 16 * kHD; i += 224) {
            int r = i / kHD, d = i - r * kHD;
            int colk = kC + h * kHD + d;
            int colv = 2 * kC + h * kHD + d;
            size_t base = ((size_t)b * kN + kt * 16 + r) * kIC;
            lK[i] = Y[base + colk] * scale1[colk] + shift1[colk];
            lV[i] = Y[base + colv] * scale1[colv] + shift1[colv];
        }
        __syncthreads();

        // S = Q @ K^T (16x16 per wave), K-depth 48
        v8f s = {};
        const int qrow = wave * 16 + lhalf;
        #pragma unroll
        for (int kk = 0; kk < kHD; kk += 4) {
            v2f a  = *(const v2f*)&lQ[qrow * kHD + kk + 2 * lhi];
            v2f bb = *(const v2f*)&lK[lhalf * kHD + kk + 2 * lhi];
            s = WMMA_F32(a, bb, s);
        }

        // + rel-pos bias (pre-gathered, coalesced in m)
        #pragma unroll
        for (int r = 0; r < 8; ++r) {
            int n = qbase + wave * 16 + r + 8 * lhi;
            s[r] += bg[(size_t)n * kN + kt * 16 + lhalf];
        }

        // online softmax: row reductions within each 16-lane half
        float f[8];
        #pragma unroll
        for (int r = 0; r < 8; ++r) {
            float v = s[r];
            #pragma unroll
            for (int off = 1; off < 16; off <<= 1)
                v = fmaxf(v, __shfl_xor(v, off, 32));
            float mnew = fmaxf(m_run[r], v);
            f[r] = __expf(m_run[r] - mnew);
            m_run[r] = mnew;
            float p = __expf(s[r] - mnew);
            s[r] = p;
            float rs = p;
            #pragma unroll
            for (int off = 1; off < 16; off <<= 1)
                rs += __shfl_xor(rs, off, 32);
            l_run[r] = l_run[r] * f[r] + rs;
        }
        #pragma unroll
        for (int r = 0; r < 8; ++r) { o0[r] *= f[r]; o1[r] *= f[r]; o2[r] *= f[r]; }

        // bounce P through LDS: C layout -> row-major -> A fragments
        float* lp = &lP[wave][0];
        #pragma unroll
        for (int r = 0; r < 8; ++r)
            lp[(r + 8 * lhi) * 16 + lhalf] = s[r];
        // same-wave LDS RAW: DS ops are in order per wave; compiler inserts dscnt waits

        // O += P @ V   (K-depth 16, three 16-wide d tiles)
        #pragma unroll
        for (int kk = 0; kk < 16; kk += 4) {
            v2f a = *(const v2f*)&lp[lhalf * 16 + kk + 2 * lhi];
            int kidx = kk + 2 * lhi;
            v2f b0, b1, b2;
            b0.x = lV[kidx * kHD + 0 + lhalf];  b0.y = lV[(kidx + 1) * kHD + 0 + lhalf];
            b1.x = lV[kidx * kHD + 16 + lhalf]; b1.y = lV[(kidx + 1) * kHD + 16 + lhalf];
            b2.x = lV[kidx * kHD + 32 + lhalf]; b2.y = lV[(kidx + 1) * kHD + 32 + lhalf];
            o0 = WMMA_F32(a, b0, o0);
            o1 = WMMA_F32(a, b1, o1);
            o2 = WMMA_F32(a, b2, o2);
        }
    }

    // epilogue: 1/l + hardswish, write [B,N,C]
    #pragma unroll
    for (int r = 0; r < 8; ++r) {
        float inv = 1.0f / l_run[r];
        int n = qbase + wave * 16 + r + 8 * lhi;
        size_t rowb = ((size_t)b * kN + n) * kC + h * kHD + lhalf;
        float x0 = o0[r] * inv, x1 = o1[r] * inv, x2 = o2[r] * inv;
        Ohs[rowb + 0]  = x0 * fminf(fmaxf(x0 + 3.f, 0.f), 6.f) * (1.f / 6.f);
        Ohs[rowb + 16] = x1 * fminf(fmaxf(x1 + 3.f, 0.f), 6.f) * (1.f / 6.f);
        Ohs[rowb + 32] = x2 * fminf(fmaxf(x2 + 3.f, 0.f), 6.f) * (1.f / 6.f);
    }
}

// ---------------------------------------------------------------------------
// Final BN apply
// ---------------------------------------------------------------------------
__global__ __launch_bounds__(256)
void bn_apply(const float* __restrict__ Z, const float* __restrict__ scale,
              const float* __restrict__ shift, float* __restrict__ out,
              int total, int nc)
{
    int i = blockIdx.x * 256 + threadIdx.x;
    if (i >= total) return;
    int c = i % nc;
    out[i] = Z[i] * scale[c] + shift[c];
}

// ---------------------------------------------------------------------------
// Launch
// ---------------------------------------------------------------------------
extern "C" void kernel_launch(void* const* d_in, const int* in_sizes, int n_in,
                              void* d_out, int out_size, void* d_ws, size_t ws_size,
                              hipStream_t stream)
{
    const float* x      = (const float*)d_in[0];
    const float* Wqkv   = (const float*)d_in[1];
    const float* g1     = (const float*)d_in[2];
    const float* b1     = (const float*)d_in[3];
    const float* Wproj  = (const float*)d_in[4];
    const float* g2     = (const float*)d_in[5];
    const float* b2     = (const float*)d_in[6];
    const float* biases = (const float*)d_in[7];
    const int*   idx    = (const int*)d_in[8];
    float* out = (float*)d_out;

    float* ws = (float*)d_ws;
    float* Y    = ws;                                   // 12544*1152
    float* Bg   = Y    + (size_t)kRows * kIC;           // 8*784*784
    float* Ohs  = Bg   + (size_t)kH * kN * kN;          // 12544*384
    float* Z    = Ohs  + (size_t)kRows * kC;            // 12544*384
    float* part = Z    + (size_t)kRows * kC;            // 32*2*1152
    float* sc1  = part + (size_t)32 * 2 * kIC;
    float* sh1  = sc1 + kIC;
    float* sc2  = sh1 + kIC;
    float* sh2  = sc2 + kC;

    const int nchunk = 32, rowsPerChunk = kRows / nchunk;  // 392
    const float invRows = 1.0f / (float)kRows;
    const float hdScale = 0.14433756729740643f;            // 48^-0.5

    // 1) QKV GEMM
    gemm_nt_f32<<<dim3(kRows / 128, kIC / 64), 256, 0, stream>>>(x, Wqkv, Y, kRows, kIC, kC);
    // 2-3) BN1 stats
    colstats_partial<<<dim3(kIC / 128, nchunk), 128, 0, stream>>>(Y, part, kRows, kIC, rowsPerChunk);
    colstats_final<<<dim3((kIC + 255) / 256), 256, 0, stream>>>(part, g1, b1, sc1, sh1, kIC, nchunk, invRows);
    // 4) bias gather expansion
    bias_expand<<<dim3(kN, kH), 256, 0, stream>>>(biases, idx, Bg);
    // 5) fused attention (+BN apply, +hardswish)
    attention_fused<<<dim3(kN / kQT, kH, kB), 224, 0, stream>>>(Y, sc1, sh1, Bg, Ohs, hdScale);
    // 6) proj GEMM
    gemm_nt_f32<<<dim3(kRows / 128, kC / 64), 256, 0, stream>>>(Ohs, Wproj, Z, kRows, kC, kC);
    // 7-8) BN2 stats
    colstats_partial<<<dim3(kC / 128, nchunk), 128, 0, stream>>>(Z, part, kRows, kC, rowsPerChunk);
    colstats_final<<<dim3((kC + 255) / 256), 256, 0, stream>>>(part, g2, b2, sc2, sh2, kC, nchunk, invRows);
    // 9) BN2 apply -> out
    bn_apply<<<dim3((kRows * kC + 255) / 256), 256, 0, stream>>>(Z, sc2, sh2, out, kRows * kC, kC);
}